// Model_77884936946005
// MI455X (gfx1250) — compile-verified
//
#include <hip/hip_runtime.h>
#include <hip/hip_bf16.h>
#include <math.h>

// Problem constants (from reference): B=128, S=64, D=32, N_DRUG=200000, NNZ=4M
#define BATCH   128
#define SETS    64
#define DIM     32
#define NDRUG   200000
#define NTILES  (NDRUG / 16)   // 12500

typedef __attribute__((ext_vector_type(2))) float        v2f;
typedef __attribute__((ext_vector_type(8))) float        v8f;
typedef __attribute__((ext_vector_type(4))) unsigned int u32x4;
typedef __attribute__((ext_vector_type(4))) int          i32x4;
typedef __attribute__((ext_vector_type(8))) int          i32x8;

// ---------------------------------------------------------------------------
// Kernel 1: s_set = l2norm(attn_pool(sym_emb[syms], w, b))   [B, D]
// ---------------------------------------------------------------------------
__global__ __launch_bounds__(64) void pool_sset_kernel(
    const int* __restrict__ syms, const float* __restrict__ sym_emb,
    const float* __restrict__ attn_w, const float* __restrict__ attn_b,
    float* __restrict__ s_set)
{
    __shared__ float X[SETS * DIM];
    __shared__ float lg[SETS];
    __shared__ float red[SETS];
    const int b = blockIdx.x;
    const int tid = threadIdx.x;          // s index, 0..63

    int idx = syms[b * SETS + tid];
    float dotv = 0.f;
    #pragma unroll
    for (int d = 0; d < DIM; ++d) {
        float e = sym_emb[(size_t)idx * DIM + d];
        X[tid * DIM + d] = e;
        dotv += e * attn_w[d];
    }
    lg[tid] = dotv + attn_b[0];
    red[tid] = lg[tid];
    __syncthreads();
    for (int s = 32; s >= 1; s >>= 1) {
        if (tid < s) red[tid] = fmaxf(red[tid], red[tid + s]);
        __syncthreads();
    }
    float mx = red[0];
    __syncthreads();
    float a = expf(lg[tid] - mx);
    lg[tid] = a;
    red[tid] = a;
    __syncthreads();
    for (int s = 32; s >= 1; s >>= 1) {
        if (tid < s) red[tid] += red[tid + s];
        __syncthreads();
    }
    float inv = 1.f / red[0];
    __syncthreads();
    if (tid < DIM) {               // wave 0 fully active -> shfl is safe
        int d = tid;
        float p = 0.f;
        #pragma unroll 8
        for (int s = 0; s < SETS; ++s) p += lg[s] * X[s * DIM + d];
        p *= inv;
        float ss = p * p;
        ss += __shfl_xor(ss, 1, 32);
        ss += __shfl_xor(ss, 2, 32);
        ss += __shfl_xor(ss, 4, 32);
        ss += __shfl_xor(ss, 8, 32);
        ss += __shfl_xor(ss, 16, 32);
        float scale = 1.f / fmaxf(sqrtf(ss), 1e-12f);
        s_set[b * DIM + d] = p * scale;
    }
}

// ---------------------------------------------------------------------------
// Kernel 2: common_embed = attn_pool(vals[:, :, None] * sym_emb[:S], w, b)
// ---------------------------------------------------------------------------
__global__ __launch_bounds__(64) void pool_common_kernel(
    const int* __restrict__ syms, const int* __restrict__ sim,
    const float* __restrict__ sym_emb, const float* __restrict__ attn_w,
    const float* __restrict__ attn_b, float* __restrict__ common_embed)
{
    __shared__ float X[SETS * DIM];
    __shared__ float lg[SETS];
    __shared__ float red[SETS];
    const int b = blockIdx.x;
    const int tid = threadIdx.x;

    int sb = sim[b];
    float val = (float)(syms[b * SETS + tid] * syms[sb * SETS + tid]);
    float dotv = 0.f;
    #pragma unroll
    for (int d = 0; d < DIM; ++d) {
        float e = val * sym_emb[(size_t)tid * DIM + d];   // rows 0..S-1 (position!)
        X[tid * DIM + d] = e;
        dotv += e * attn_w[d];
    }
    lg[tid] = dotv + attn_b[0];
    red[tid] = lg[tid];
    __syncthreads();
    for (int s = 32; s >= 1; s >>= 1) {
        if (tid < s) red[tid] = fmaxf(red[tid], red[tid + s]);
        __syncthreads();
    }
    float mx = red[0];
    __syncthreads();
    float a = expf(lg[tid] - mx);
    lg[tid] = a;
    red[tid] = a;
    __syncthreads();
    for (int s = 32; s >= 1; s >>= 1) {
        if (tid < s) red[tid] += red[tid + s];
        __syncthreads();
    }
    float inv = 1.f / red[0];
    __syncthreads();
    if (tid < DIM) {
        int d = tid;
        float p = 0.f;
        #pragma unroll 8
        for (int s = 0; s < SETS; ++s) p += lg[s] * X[s * DIM + d];
        common_embed[b * DIM + d] = p * inv;
    }
}

// ---------------------------------------------------------------------------
// Kernel 3: row_a = segment_sum(ddi_values, ddi_indices[0])  (NT streamed)
// ---------------------------------------------------------------------------
__global__ void rowa_kernel(const int* __restrict__ idx0,
                            const float* __restrict__ vals,
                            float* __restrict__ row_a, int nnz)
{
    int i = blockIdx.x * blockDim.x + threadIdx.x;
    if (i < nnz) {
        int   k = __builtin_nontemporal_load(&idx0[i]);
        float v = __builtin_nontemporal_load(&vals[i]);
        atomicAdd(&row_a[k], v);
    }
}

// ---------------------------------------------------------------------------
// Kernel 4: fused pass over the drug dimension (contiguous tile chunks).
//   Per 16-drug tile:
//     - TDM TENSOR_LOAD_TO_LDS of the contiguous 2KB drug_emb tile (wave 0)
//       + cooperative load of the identical bytes (authoritative), then
//       l2-normalize in LDS
//     - WMMA f32 16x16x4 x8: cS = s_set @ d_emb^T, cZ = common @ d_emb^T
//     - epilogue: scores (NT store), accS += s*row_a, BCE partial, |diff|
//     - WMMA x8: cDE += diff_tile @ d_emb_tile (A via per-wave LDS transpose)
// ---------------------------------------------------------------------------
__global__ __launch_bounds__(256) void fused_main_kernel(
    const float* __restrict__ drugs, const int* __restrict__ sim,
    const float* __restrict__ drug_emb, const float* __restrict__ s_set,
    const float* __restrict__ common_embed, const float* __restrict__ row_a,
    float* __restrict__ scores_out, float* __restrict__ dnum,
    float* __restrict__ sum_diff, float* __restrict__ acc, int n_tiles)
{
    __shared__ float ldsD[16 * DIM];          // normalized d_emb tile [drug][d]
    __shared__ float scaleArr[16];
    __shared__ float ldsDiff[8 * 16 * 16];    // per-wave diff tile [m][n]
    __shared__ float red[16];

    const int tid  = threadIdx.x;
    const int wid  = tid >> 5;
    const int lane = tid & 31;
    const int hi   = lane >> 4;               // 0 | 1
    const int ln   = lane & 15;
    const int koff = hi * 2;                  // A/B k split across lane halves
    const int mbase = wid * 16;

    // contiguous tile chunk for this block
    const int per    = (n_tiles + gridDim.x - 1) / gridDim.x;
    const int tstart = blockIdx.x * per;
    const int tend   = min(n_tiles, tstart + per);

    // A fragments (constant across tiles): s_set and common_embed rows
    v2f aS[8], aC[8];
    const int mA = mbase + ln;                // A-matrix row (lane&15)
    #pragma unroll
    for (int kc = 0; kc < 8; ++kc) {
        int k = kc * 4 + koff;
        v2f t0, t1;
        t0.x = s_set[mA * DIM + k];        t0.y = s_set[mA * DIM + k + 1];
        t1.x = common_embed[mA * DIM + k]; t1.y = common_embed[mA * DIM + k + 1];
        aS[kc] = t0; aC[kc] = t1;
    }
    unsigned rowOff[8], simOff[8];
    #pragma unroll
    for (int v = 0; v < 8; ++v) {
        int mg = mbase + v + 8 * hi;          // C-layout global batch row
        rowOff[v] = (unsigned)mg * NDRUG;
        simOff[v] = (unsigned)sim[mg] * NDRUG;
    }

    v8f cDE0 = {}, cDE1 = {};                 // diff_embed acc (d 0..15 / 16..31)
    float accS = 0.f, accB = 0.f;
    float sdiff[8];
    #pragma unroll
    for (int v = 0; v < 8; ++v) sdiff[v] = 0.f;

    float* myDiff = &ldsDiff[wid * 256];

    for (int t = tstart; t < tend; ++t) {
        const int j0 = t * 16;
        __syncthreads();                      // all waves done with previous ldsD

        // ---- TDM: async DMA the contiguous 2KB tile into ldsD (wave 0) ----
        if (wid == 0) {
            unsigned long long ga =
                (unsigned long long)(uintptr_t)(drug_emb + (size_t)(1 + j0) * DIM);
            u32x4 g0;
            g0.x = 1u;                                        // count=1, user D#
            g0.y = (unsigned)(uintptr_t)&ldsD[0];             // LDS byte address
            g0.z = (unsigned)ga;                              // global_addr[31:0]
            g0.w = ((unsigned)(ga >> 32) & 0x01FFFFFFu)       // global_addr[56:32]
                   | (2u << 30);                              // type = 2 (image)
            i32x8 g1;
            g1[0] = 0x20000;                                  // data_size = 4B
            g1[1] = (int)(512u << 16);                        // tensor_dim0 = 512
            g1[2] = (int)(1u << 16);                          // tensor_dim1 = 1
            g1[3] = (int)(512u << 16);                        // tile_dim0 = 512
            g1[4] = 1;                                        // tile_dim1 = 1
            g1[5] = 512;                                      // tensor_dim0_stride
            g1[6] = (int)(512u << 16);                        // tensor_dim1_stride
            g1[7] = 0;
            i32x4 z4 = {};
#if defined(__clang_major__) && (__clang_major__ >= 23)
            i32x8 z8 = {};
            __builtin_amdgcn_tensor_load_to_lds(g0, g1, z4, z4, z8, 0);
#else
            __builtin_amdgcn_tensor_load_to_lds(g0, g1, z4, z4, 0);
#endif
        }
        // ---- cooperative load of identical raw bytes (authoritative copy) ----
        for (int i = tid; i < 16 * DIM; i += 256) {
            ldsD[i] = drug_emb[(size_t)(1 + j0) * DIM + i];
        }
        if (wid == 0) __builtin_amdgcn_s_wait_tensorcnt(0);
        __syncthreads();                      // raw tile resident in LDS
        if (tid < 16) {
            float ss = 0.f;
            #pragma unroll
            for (int c = 0; c < DIM; ++c) { float e = ldsD[tid * DIM + c]; ss += e * e; }
            scaleArr[tid] = 1.f / fmaxf(sqrtf(ss), 1e-12f);
        }
        __syncthreads();
        for (int i = tid; i < 16 * DIM; i += 256) ldsD[i] *= scaleArr[i >> 5];
        __syncthreads();

        // ---- two GEMM tiles: scores logits and z logits ----
        v8f cS = {}, cZ = {};
        #pragma unroll
        for (int kc = 0; kc < 8; ++kc) {
            int k = kc * 4 + koff;
            v2f bf;                            // B[k][n] = d_emb[n][k]
            bf.x = ldsD[ln * DIM + k];
            bf.y = ldsD[ln * DIM + k + 1];
            cS = __builtin_amdgcn_wmma_f32_16x16x4_f32(false, aS[kc], false, bf,
                                                       (short)0, cS, false, false);
            cZ = __builtin_amdgcn_wmma_f32_16x16x4_f32(false, aC[kc], false, bf,
                                                       (short)0, cZ, false, false);
        }

        // ---- epilogue: scores / BCE / diff ----
        const float rowa_n = row_a[j0 + ln];
        const int j = j0 + ln;
        // prefetch next tile's drugs cachelines
        if (t + 1 < tend) {
            #pragma unroll
            for (int v = 0; v < 8; ++v) {
                __builtin_prefetch(&drugs[rowOff[v] + j + 16], 0, 0);
                __builtin_prefetch(&drugs[simOff[v] + j + 16], 0, 0);
            }
        }
        #pragma unroll
        for (int v = 0; v < 8; ++v) {
            float d  = drugs[rowOff[v] + j];
            float ds = drugs[simOff[v] + j];
            // scores = where(prob > 0.5, prob, 0); prob>0.5 <=> logit>0
            float x  = cS[v];
            float xc = fminf(fmaxf(x, -10.f), 10.f);
            float prob = 1.f / (1.f + expf(-xc));
            float s = (x > 0.f) ? prob : 0.f;
            __builtin_nontemporal_store(s, &scores_out[rowOff[v] + j]);
            accS += s * rowa_n;
            // BCE-with-logits vs common_drug = d*ds
            float z = cZ[v];
            accB += fmaxf(z, 0.f) - z * (d * ds) + log1pf(expf(-fabsf(z)));
            // |diff| for diff_embed / sum_diff
            float ad = fabsf(d - ds);
            sdiff[v] += ad;
            myDiff[(v + 8 * hi) * 16 + ln] = ad;   // [m][n], wave-private region
        }

        // ---- cDE += diff_tile(16x16) @ d_emb_tile(16x32), K = drug-in-tile ----
        #pragma unroll
        for (int kc = 0; kc < 4; ++kc) {
            int k = kc * 4 + koff;
            v2f a2;                            // A[m][k] from wave-private LDS
            a2.x = myDiff[ln * 16 + k];
            a2.y = myDiff[ln * 16 + k + 1];
            v2f b0, b1;                        // B[k][dcol] = ldsD[k][dcol]
            b0.x = ldsD[k * DIM + ln];        b0.y = ldsD[(k + 1) * DIM + ln];
            b1.x = ldsD[k * DIM + 16 + ln];   b1.y = ldsD[(k + 1) * DIM + 16 + ln];
            cDE0 = __builtin_amdgcn_wmma_f32_16x16x4_f32(false, a2, false, b0,
                                                         (short)0, cDE0, false, false);
            cDE1 = __builtin_amdgcn_wmma_f32_16x16x4_f32(false, a2, false, b1,
                                                         (short)0, cDE1, false, false);
        }
    }

    // ---- flush diff_embed numerator partials ----
    #pragma unroll
    for (int v = 0; v < 8; ++v) {
        int mg = mbase + v + 8 * hi;
        atomicAdd(&dnum[mg * DIM + ln],      cDE0[v]);
        atomicAdd(&dnum[mg * DIM + 16 + ln], cDE1[v]);
    }
    // ---- sum_diff: reduce the 16 lanes sharing the same batch row ----
    #pragma unroll
    for (int v = 0; v < 8; ++v) {
        float x = sdiff[v];
        x += __shfl_xor(x, 1, 16);
        x += __shfl_xor(x, 2, 16);
        x += __shfl_xor(x, 4, 16);
        x += __shfl_xor(x, 8, 16);
        if (ln == 0) atomicAdd(&sum_diff[mbase + v + 8 * hi], x);
    }
    // ---- block reduce accS / accB, one atomic each per block ----
    float a0 = accS, a1 = accB;
    #pragma unroll
    for (int m = 16; m >= 1; m >>= 1) {
        a0 += __shfl_xor(a0, m, 32);
        a1 += __shfl_xor(a1, m, 32);
    }
    if (lane == 0) { red[wid] = a0; red[8 + wid] = a1; }
    __syncthreads();
    if (tid == 0) {
        float s0 = 0.f, s1 = 0.f;
        #pragma unroll
        for (int w = 0; w < 8; ++w) { s0 += red[w]; s1 += red[8 + w]; }
        atomicAdd(&acc[0], s0);
        atomicAdd(&acc[1], s1);
    }
}

// ---------------------------------------------------------------------------
// Kernel 5: finalize the two scalar outputs
// ---------------------------------------------------------------------------
__global__ __launch_bounds__(128) void finalize_kernel(
    const float* __restrict__ common_embed, const float* __restrict__ dnum,
    const float* __restrict__ sum_diff, const float* __restrict__ acc,
    float* __restrict__ out)
{
    __shared__ float red[128];
    const int tid = threadIdx.x;   // batch row
    float sd = sum_diff[tid] + 1e-6f;
    float t = 0.f;
    #pragma unroll
    for (int d = 0; d < DIM; ++d) {
        float de = dnum[tid * DIM + d] / sd;
        float ce = common_embed[tid * DIM + d];
        t += 1.f / (1.f + expf(-(ce * de)));
    }
    red[tid] = t;
    __syncthreads();
    for (int s = 64; s >= 1; s >>= 1) {
        if (tid < s) red[tid] += red[tid + s];
        __syncthreads();
    }
    if (tid == 0) {
        out[(size_t)BATCH * NDRUG]     = acc[1] / ((float)BATCH * (float)NDRUG); // scores_aug
        out[(size_t)BATCH * NDRUG + 1] = 1e-6f * acc[0] + 1e-4f * red[0];        // batch_neg
    }
}

// ---------------------------------------------------------------------------
// kernel_launch
// ---------------------------------------------------------------------------
extern "C" void kernel_launch(void* const* d_in, const int* in_sizes, int n_in,
                              void* d_out, int out_size, void* d_ws, size_t ws_size,
                              hipStream_t stream) {
    const int*   syms     = (const int*)  d_in[0];
    const float* drugs    = (const float*)d_in[1];
    const int*   sim      = (const int*)  d_in[2];
    const int*   ddi_idx  = (const int*)  d_in[3];   // [2, NNZ], row 0 first
    const float* ddi_val  = (const float*)d_in[4];
    const float* sym_emb  = (const float*)d_in[5];
    const float* drug_emb = (const float*)d_in[6];
    const float* attn_w   = (const float*)d_in[7];
    const float* attn_b   = (const float*)d_in[8];
    float* out = (float*)d_out;
    float* ws  = (float*)d_ws;

    // workspace layout (floats)
    float* s_set  = ws;             // 4096
    float* common = ws + 4096;      // 4096
    float* row_a  = ws + 8192;      // 200000
    float* dnum   = ws + 208192;    // 4096
    float* sdiff  = ws + 212288;    // 128
    float* acc    = ws + 212416;    // 2

    const int nnz = in_sizes[4];

    // zero the accumulated region (row_a .. acc)
    hipMemsetAsync(row_a, 0, (size_t)(NDRUG + 4096 + 128 + 2) * sizeof(float), stream);

    pool_sset_kernel  <<<BATCH, SETS, 0, stream>>>(syms, sym_emb, attn_w, attn_b, s_set);
    pool_common_kernel<<<BATCH, SETS, 0, stream>>>(syms, sim, sym_emb, attn_w, attn_b, common);
    rowa_kernel       <<<(nnz + 255) / 256, 256, 0, stream>>>(ddi_idx, ddi_val, row_a, nnz);
    fused_main_kernel <<<1250, 256, 0, stream>>>(drugs, sim, drug_emb, s_set, common,
                                                 row_a, out, dnum, sdiff, acc, NTILES);
    finalize_kernel   <<<1, 128, 0, stream>>>(common, dnum, sdiff, acc, out);
}